// SetConv2dEncoder_75943611728142
// MI455X (gfx1250) — compile-verified
//
#include <hip/hip_runtime.h>
#include <cstddef>

typedef __attribute__((ext_vector_type(2))) float v2f;
typedef __attribute__((ext_vector_type(8))) float v8f;

#define B_   4
#define N_   1024
#define C_   16
#define M_   1024
#define EPS_ 1e-8f
#define NT   64      // n-chunk depth staged in LDS
#define LDW  48      // padded LDS row stride (2-row step -> +96 = 32 mod 64 banks)

__global__ __launch_bounds__(256) void setconv2d_fused(
    const float* __restrict__ xz,        // (B, N, 1)
    const float* __restrict__ z,         // (B, N, C)
    const float* __restrict__ xg,        // (M)
    const float* __restrict__ log_scale, // (1)
    float* __restrict__ out)             // (B, 18, M, M)
{
    __shared__ float Wi[NT][LDW];
    __shared__ float Wj[NT][LDW];
    __shared__ float Zc[NT][17];

    const int b     = blockIdx.z;
    const int iBase = blockIdx.y * 32;
    const int jBase = blockIdx.x * 32;
    const int tid   = threadIdx.x;
    const int lane  = tid & 31;
    const int wid   = tid >> 5;
    const int iSub  = (wid >> 1) & 1;   // which 16-row half of the 32x32 patch
    const int jSub  = wid & 1;          // which 16-col half
    const int grp   = wid >> 2;         // channel group: k in [1..8] or [9..16]

    const int h = lane >> 4;            // lane half
    const int r = lane & 15;

    // weights = exp(-0.5*d^2 / exp(2*ls)) = exp(sc*d^2), sc = -0.5*exp(-2*ls)
    const float sc = -0.5f * expf(-2.0f * log_scale[0]);

    const float* xzb = xz + b * N_;
    const float* zb  = z  + (size_t)b * N_ * C_;

    v8f acc[9];
    #pragma unroll
    for (int q = 0; q < 9; ++q)
        #pragma unroll
        for (int v = 0; v < 8; ++v)
            acc[q][v] = 0.0f;

    const int icol = iSub * 16 + r;
    const int jcol = jSub * 16 + r;

    for (int n0 = 0; n0 < N_; n0 += NT) {
        __syncthreads();   // previous chunk's LDS reads done

        // ---- stage Gaussian weights for this n-chunk (computed on the fly) ----
        #pragma unroll
        for (int e = tid; e < NT * 32; e += 256) {
            const int nn  = e >> 5;
            const int col = e & 31;
            const float xv = xzb[n0 + nn];
            const float di = xv - xg[iBase + col];
            const float dj = xv - xg[jBase + col];
            Wi[nn][col] = expf(sc * di * di);
            Wj[nn][col] = expf(sc * dj * dj);
        }
        // ---- stage zc = [1, z] ----
        if (tid < NT) Zc[tid][0] = 1.0f;
        #pragma unroll
        for (int e = tid; e < NT * C_; e += 256) {
            const int nn = e >> 4;
            const int c  = e & 15;
            Zc[nn][1 + c] = zb[(size_t)(n0 + nn) * C_ + c];
        }
        if (n0 + NT < N_) {
            __builtin_prefetch(&zb[(size_t)(n0 + NT) * C_], 0, 0);  // global_prefetch_b8
        }
        __syncthreads();

        // ---- 16 K-steps of V_WMMA_F32_16X16X4_F32 per wave ----
        #pragma unroll 4
        for (int kk = 0; kk < NT; kk += 4) {
            const int na = kk + 2 * h;   // A/B fragment rows: K = 2*h + v
            v2f a, bb;
            a.x  = Wi[na][icol];     // A[M=r][K]   = W[n, i]  (transposed access)
            a.y  = Wi[na + 1][icol];
            bb.x = Wj[na][jcol];     // B[K][N=r]   = W[n, j]
            bb.y = Wj[na + 1][jcol];

            // k = 0 (density): zc == 1
            acc[0] = __builtin_amdgcn_wmma_f32_16x16x4_f32(
                false, a, false, bb, (short)0, acc[0], false, false);

            const float zc0a = Zc[na][0];      // keep Zc row addresses warm (no-op value)
            (void)zc0a;
            #pragma unroll
            for (int q = 1; q <= 8; ++q) {
                const int k = grp * 8 + q;     // uniform per wave
                v2f bk;
                bk.x = bb.x * Zc[na][k];       // row-scale B by zc[n, k]
                bk.y = bb.y * Zc[na + 1][k];
                acc[q] = __builtin_amdgcn_wmma_f32_16x16x4_f32(
                    false, a, false, bk, (short)0, acc[q], false, false);
            }
        }
    }

    // ---- epilogue: normalization + all stores, fused ----
    const int iRow0 = iBase + iSub * 16 + 8 * h;   // C/D: M = v + 8*h
    const int jCol  = jBase + jSub * 16 + r;       //      N = r
    float* ob = out + (size_t)b * 18 * M_ * M_;
    const size_t chS = (size_t)M_ * M_;

    v8f dens = acc[0];
    v8f invd;
    #pragma unroll
    for (int v = 0; v < 8; ++v) invd[v] = 1.0f / (dens[v] + EPS_);

    if (grp == 0) {
        // channel 1: raw density
        #pragma unroll
        for (int v = 0; v < 8; ++v)
            ob[chS + (size_t)(iRow0 + v) * M_ + jCol] = dens[v];
    } else {
        // channel 0: identity
        #pragma unroll
        for (int v = 0; v < 8; ++v)
            ob[(size_t)(iRow0 + v) * M_ + jCol] = ((iRow0 + v) == jCol) ? 1.0f : 0.0f;
    }
    #pragma unroll
    for (int q = 1; q <= 8; ++q) {
        const int ch = 1 + grp * 8 + q;   // channels 2..17
        #pragma unroll
        for (int v = 0; v < 8; ++v)
            ob[(size_t)ch * chS + (size_t)(iRow0 + v) * M_ + jCol] = acc[q][v] * invd[v];
    }
}

extern "C" void kernel_launch(void* const* d_in, const int* in_sizes, int n_in,
                              void* d_out, int out_size, void* d_ws, size_t ws_size,
                              hipStream_t stream) {
    const float* xz = (const float*)d_in[0];
    const float* z  = (const float*)d_in[1];
    const float* xg = (const float*)d_in[2];
    const float* ls = (const float*)d_in[3];
    float* out = (float*)d_out;

    // first output of the tuple: x_grid itself
    hipMemcpyAsync(out, xg, M_ * sizeof(float), hipMemcpyDeviceToDevice, stream);

    dim3 grid(M_ / 32, M_ / 32, B_);
    setconv2d_fused<<<grid, 256, 0, stream>>>(xz, z, xg, ls, out + M_);
}